// DynamicGNN_77730318123062
// MI455X (gfx1250) — compile-verified
//
#include <hip/hip_runtime.h>
#include <hip/hip_bf16.h>
#include <math.h>

#define D_CH   128
#define OUT_CH 64

typedef __attribute__((ext_vector_type(2))) float v2f;
typedef __attribute__((ext_vector_type(8))) float v8f;

// ---------------------------------------------------------------- utilities
__device__ __forceinline__ float leaky02(float x) { return x > 0.0f ? x : 0.2f * x; }

__device__ __forceinline__ void atomicMaxF(float* addr, float val) {
    unsigned int* u = (unsigned int*)addr;
    unsigned int cur = *((volatile unsigned int*)u);
    while (__uint_as_float(cur) < val) {
        unsigned int prev = atomicCAS(u, cur, __float_as_uint(val));
        if (prev == cur) break;
        cur = prev;
    }
}

// ---------------------------------------------------------------- elementwise
__global__ void fill_kernel(float* __restrict__ p, float v, int n) {
    int i = blockIdx.x * blockDim.x + threadIdx.x;
    if (i < n) p[i] = v;
}

__global__ void deg_scatter_kernel(const int* __restrict__ dst, float* __restrict__ deg, int E) {
    int e = blockIdx.x * blockDim.x + threadIdx.x;
    if (e < E) atomicAdd(&deg[dst[e]], 1.0f);
}

__global__ void rsqrt_kernel(float* __restrict__ dinv, const float* __restrict__ deg, int n) {
    int i = blockIdx.x * blockDim.x + threadIdx.x;
    if (i < n) {
        float d = deg[i];
        dinv[i] = d > 0.0f ? rsqrtf(d) : 0.0f;
    }
}

// out[i,c] = h[i,c] * dinv[i]^2   (GCN self-loop contribution, also clears poison)
__global__ void gcn_self_init_kernel(float* __restrict__ out, const float* __restrict__ h,
                                     const float* __restrict__ dinv, int n, int C) {
    int idx = blockIdx.x * blockDim.x + threadIdx.x;
    if (idx >= n * C) return;
    int i = idx / C;
    float w = dinv[i] * dinv[i];
    out[idx] = h[idx] * w;
}

__global__ void bias_act_kernel(float* __restrict__ io, const float* __restrict__ b,
                                int n, int C, int do_relu) {
    int idx = blockIdx.x * blockDim.x + threadIdx.x;
    if (idx >= n * C) return;
    float v = io[idx] + b[idx % C];
    io[idx] = do_relu ? fmaxf(v, 0.0f) : v;
}

__global__ void mean_div_kernel(float* __restrict__ agg, const float* __restrict__ deg,
                                int n, int C) {
    int idx = blockIdx.x * blockDim.x + threadIdx.x;
    if (idx >= n * C) return;
    int i = idx / C;
    agg[idx] = agg[idx] / fmaxf(deg[i], 1.0f);
}

// ---------------------------------------------------------------- edge scatters (warp per edge)
__global__ void gcn_edge_kernel(float* __restrict__ out, const float* __restrict__ h,
                                const float* __restrict__ dinv,
                                const int* __restrict__ src, const int* __restrict__ dst,
                                int E, int C) {
    int gid = blockIdx.x * blockDim.x + threadIdx.x;
    int e = gid >> 5;
    if (e >= E) return;
    int lane = gid & 31;
    int s = src[e], d = dst[e];
    float w = dinv[s] * dinv[d];
    const float* hs = h + (size_t)s * C;
    float* od = out + (size_t)d * C;
    for (int c = lane; c < C; c += 32)
        atomicAdd(&od[c], hs[c] * w);
}

__global__ void sage_edge_kernel(float* __restrict__ agg, const float* __restrict__ h,
                                 const int* __restrict__ src, const int* __restrict__ dst,
                                 int E, int C) {
    int gid = blockIdx.x * blockDim.x + threadIdx.x;
    int e = gid >> 5;
    if (e >= E) return;
    int lane = gid & 31;
    int s = src[e], d = dst[e];
    const float* hs = h + (size_t)s * C;
    float* od = agg + (size_t)d * C;
    for (int c = lane; c < C; c += 32)
        atomicAdd(&od[c], hs[c]);
}

// ---------------------------------------------------------------- GAT pieces
__global__ void alpha_kernel(const float* __restrict__ hg, const float* __restrict__ a_s,
                             const float* __restrict__ a_d, float* __restrict__ as_,
                             float* __restrict__ ad_, int n, int C) {
    int i = blockIdx.x * blockDim.x + threadIdx.x;
    if (i >= n) return;
    const float* row = hg + (size_t)i * C;
    float s = 0.0f, d = 0.0f;
    for (int c = 0; c < C; ++c) {
        float v = row[c];
        s += v * a_s[c];
        d += v * a_d[c];
    }
    as_[i] = s;
    ad_[i] = d;
}

__global__ void gat_m_init_kernel(float* __restrict__ m, const float* __restrict__ as_,
                                  const float* __restrict__ ad_, int n) {
    int i = blockIdx.x * blockDim.x + threadIdx.x;
    if (i < n) m[i] = leaky02(as_[i] + ad_[i]);   // self-loop edge
}

__global__ void gat_m_edge_kernel(float* __restrict__ m, const float* __restrict__ as_,
                                  const float* __restrict__ ad_,
                                  const int* __restrict__ src, const int* __restrict__ dst, int E) {
    int e = blockIdx.x * blockDim.x + threadIdx.x;
    if (e >= E) return;
    int s = src[e], d = dst[e];
    atomicMaxF(&m[d], leaky02(as_[s] + ad_[d]));
}

__global__ void gat_den_init_kernel(float* __restrict__ den, const float* __restrict__ as_,
                                    const float* __restrict__ ad_, const float* __restrict__ m,
                                    int n) {
    int i = blockIdx.x * blockDim.x + threadIdx.x;
    if (i < n) den[i] = __expf(leaky02(as_[i] + ad_[i]) - m[i]);
}

__global__ void gat_den_edge_kernel(float* __restrict__ den, const float* __restrict__ as_,
                                    const float* __restrict__ ad_, const float* __restrict__ m,
                                    const int* __restrict__ src, const int* __restrict__ dst, int E) {
    int e = blockIdx.x * blockDim.x + threadIdx.x;
    if (e >= E) return;
    int s = src[e], d = dst[e];
    atomicAdd(&den[d], __expf(leaky02(as_[s] + ad_[d]) - m[d]));
}

__global__ void gat_out_init_kernel(float* __restrict__ out, const float* __restrict__ hg,
                                    const float* __restrict__ as_, const float* __restrict__ ad_,
                                    const float* __restrict__ m, const float* __restrict__ den,
                                    int n, int C) {
    int idx = blockIdx.x * blockDim.x + threadIdx.x;
    if (idx >= n * C) return;
    int i = idx / C;
    float coef = __expf(leaky02(as_[i] + ad_[i]) - m[i]) / den[i];
    out[idx] = coef * hg[idx];
}

__global__ void gat_out_edge_kernel(float* __restrict__ out, const float* __restrict__ hg,
                                    const float* __restrict__ as_, const float* __restrict__ ad_,
                                    const float* __restrict__ m, const float* __restrict__ den,
                                    const int* __restrict__ src, const int* __restrict__ dst,
                                    int E, int C) {
    int gid = blockIdx.x * blockDim.x + threadIdx.x;
    int e = gid >> 5;
    if (e >= E) return;
    int lane = gid & 31;
    int s = src[e], d = dst[e];
    float coef = __expf(leaky02(as_[s] + ad_[d]) - m[d]) / den[d];
    const float* hs = hg + (size_t)s * C;
    float* od = out + (size_t)d * C;
    for (int c = lane; c < C; c += 32)
        atomicAdd(&od[c], coef * hs[c]);
}

// ---------------------------------------------------------------- WMMA fp32 GEMM
// C[M,Nc] (+)= A[M,K] @ B[K,Nc], fp32, each wave computes a 16x64 C strip.
// Uses V_WMMA_F32_16X16X4_F32 (wave32 layouts per CDNA5 ISA 7.12.2):
//   A 16x4 frag : row M = lane&15, K = k0 + (lane>>4)*2 + {0,1}  (2 VGPRs)
//   B 4x16 frag : col N = lane&15, K = k0 + (lane>>4)*2 + {0,1}  (2 VGPRs)
//   C 16x16     : col N = lane&15, row M = (lane>>4)*8 + v       (8 VGPRs)
__global__ void gemm_wmma_f32_kernel(const float* __restrict__ A, const float* __restrict__ B,
                                     float* __restrict__ C, int M, int K, int Nc,
                                     int accumulate) {
    int wave = (blockIdx.x * blockDim.x + threadIdx.x) >> 5;
    int lane = threadIdx.x & 31;
    int stripsPerRow = Nc >> 6;                 // number of 64-wide strips
    int tileM = (wave / stripsPerRow) * 16;
    int tileN = (wave % stripsPerRow) * 64;
    if (tileM >= M) return;                     // wave-uniform: EXEC stays all-ones

    int lhalf = lane >> 4;                      // 0 or 1
    int l15   = lane & 15;

    v8f acc[4];
    if (accumulate) {
        for (int t = 0; t < 4; ++t)
            for (int v = 0; v < 8; ++v)
                acc[t][v] = C[(size_t)(tileM + lhalf * 8 + v) * Nc + tileN + t * 16 + l15];
    } else {
        for (int t = 0; t < 4; ++t) {
            v8f z = {};
            acc[t] = z;
        }
    }

    const float* Arow = A + (size_t)(tileM + l15) * K;
    for (int k0 = 0; k0 < K; k0 += 4) {
        int ka = k0 + lhalf * 2;
        v2f a;
        a.x = Arow[ka + 0];
        a.y = Arow[ka + 1];
        const float* B0 = B + (size_t)(ka + 0) * Nc + tileN + l15;
        const float* B1 = B + (size_t)(ka + 1) * Nc + tileN + l15;
        for (int t = 0; t < 4; ++t) {
            v2f b;
            b.x = B0[t * 16];
            b.y = B1[t * 16];
            acc[t] = __builtin_amdgcn_wmma_f32_16x16x4_f32(
                /*neg_a=*/false, a, /*neg_b=*/false, b,
                /*c_mod=*/(short)0, acc[t], /*reuse_a=*/false, /*reuse_b=*/false);
        }
    }

    for (int t = 0; t < 4; ++t)
        for (int v = 0; v < 8; ++v)
            C[(size_t)(tileM + lhalf * 8 + v) * Nc + tileN + t * 16 + l15] = acc[t][v];
}

// ---------------------------------------------------------------- host side
static inline int ceildiv(long long a, long long b) { return (int)((a + b - 1) / b); }

static void launch_gemm(const float* A, const float* B, float* C, int M, int K, int Nc,
                        int accumulate, hipStream_t stream) {
    long long waves = (long long)(M / 16) * (Nc / 64);
    int blocks = ceildiv(waves * 32, 256);
    gemm_wmma_f32_kernel<<<blocks, 256, 0, stream>>>(A, B, C, M, K, Nc, accumulate);
}

extern "C" void kernel_launch(void* const* d_in, const int* in_sizes, int n_in,
                              void* d_out, int out_size, void* d_ws, size_t ws_size,
                              hipStream_t stream) {
    const float* x   = (const float*)d_in[0];
    const int*   ei  = (const int*)d_in[1];
    const float* W1  = (const float*)d_in[2];
    const float* b1  = (const float*)d_in[3];
    const float* Wl  = (const float*)d_in[4];
    const float* Wr  = (const float*)d_in[5];
    const float* bs  = (const float*)d_in[6];
    const float* Wg  = (const float*)d_in[7];
    const float* a_s = (const float*)d_in[8];
    const float* a_d = (const float*)d_in[9];
    const float* bg  = (const float*)d_in[10];
    const float* Wo  = (const float*)d_in[11];
    const float* bo  = (const float*)d_in[12];

    const int N = in_sizes[0] / D_CH;
    const int E = in_sizes[1] / 2;
    const int* src = ei;
    const int* dst = ei + E;

    size_t nd = (size_t)N * D_CH;
    float* ws   = (float*)d_ws;
    float* bufA = ws;                 // gemm tmp / SAGE agg-mean / GAT hg
    float* bufB = ws + nd;            // SAGE result -> h2 / layer4 h4
    float* bufC = ws + 2 * nd;        // GCN1 acc -> h1 / GAT acc -> h3
    float* vecs = ws + 3 * nd;
    float* dinv = vecs;               // N (kept for layer 4)
    float* deg  = vecs + N;           // N (reused GCN then SAGE)
    float* as_  = vecs + 2 * (size_t)N;
    float* ad_  = vecs + 3 * (size_t)N;
    float* m_   = vecs + 4 * (size_t)N;
    float* den_ = vecs + 5 * (size_t)N;
    float* out  = (float*)d_out;

    const int T = 256;
    int bN   = ceildiv(N, T);
    int bE   = ceildiv(E, T);
    int bEW  = ceildiv((long long)E * 32, T);   // warp per edge
    int bND  = ceildiv(nd, T);
    int bNO  = ceildiv((long long)N * OUT_CH, T);

    // ---------------- Layer 1: GCN + ReLU -> h1 (bufC)
    fill_kernel<<<bN, T, 0, stream>>>(deg, 1.0f, N);                      // self-loops
    deg_scatter_kernel<<<bE, T, 0, stream>>>(dst, deg, E);
    rsqrt_kernel<<<bN, T, 0, stream>>>(dinv, deg, N);
    launch_gemm(x, W1, bufA, N, D_CH, D_CH, 0, stream);                   // h = x @ W1
    gcn_self_init_kernel<<<bND, T, 0, stream>>>(bufC, bufA, dinv, N, D_CH);
    gcn_edge_kernel<<<bEW, T, 0, stream>>>(bufC, bufA, dinv, src, dst, E, D_CH);
    bias_act_kernel<<<bND, T, 0, stream>>>(bufC, b1, N, D_CH, 1);

    // ---------------- Layer 2: SAGE(mean) + ReLU -> h2 (bufB)
    fill_kernel<<<bN, T, 0, stream>>>(deg, 0.0f, N);                      // no self-loops
    deg_scatter_kernel<<<bE, T, 0, stream>>>(dst, deg, E);
    fill_kernel<<<bND, T, 0, stream>>>(bufA, 0.0f, (int)nd);
    sage_edge_kernel<<<bEW, T, 0, stream>>>(bufA, bufC, src, dst, E, D_CH);
    mean_div_kernel<<<bND, T, 0, stream>>>(bufA, deg, N, D_CH);
    launch_gemm(bufA, Wl, bufB, N, D_CH, D_CH, 0, stream);                // mean @ Wl
    launch_gemm(bufC, Wr, bufB, N, D_CH, D_CH, 1, stream);                // += h1 @ Wr
    bias_act_kernel<<<bND, T, 0, stream>>>(bufB, bs, N, D_CH, 1);

    // ---------------- Layer 3: GAT (1 head) + ReLU -> h3 (bufC)
    launch_gemm(bufB, Wg, bufA, N, D_CH, D_CH, 0, stream);                // hg = h2 @ Wg
    alpha_kernel<<<bN, T, 0, stream>>>(bufA, a_s, a_d, as_, ad_, N, D_CH);
    gat_m_init_kernel<<<bN, T, 0, stream>>>(m_, as_, ad_, N);
    gat_m_edge_kernel<<<bE, T, 0, stream>>>(m_, as_, ad_, src, dst, E);
    gat_den_init_kernel<<<bN, T, 0, stream>>>(den_, as_, ad_, m_, N);
    gat_den_edge_kernel<<<bE, T, 0, stream>>>(den_, as_, ad_, m_, src, dst, E);
    gat_out_init_kernel<<<bND, T, 0, stream>>>(bufC, bufA, as_, ad_, m_, den_, N, D_CH);
    gat_out_edge_kernel<<<bEW, T, 0, stream>>>(bufC, bufA, as_, ad_, m_, den_, src, dst, E, D_CH);
    bias_act_kernel<<<bND, T, 0, stream>>>(bufC, bg, N, D_CH, 1);

    // ---------------- Layer 4: GCN output (no ReLU) -> d_out
    launch_gemm(bufC, Wo, bufB, N, D_CH, OUT_CH, 0, stream);              // h4 = h3 @ Wo
    gcn_self_init_kernel<<<bNO, T, 0, stream>>>(out, bufB, dinv, N, OUT_CH);
    gcn_edge_kernel<<<bEW, T, 0, stream>>>(out, bufB, dinv, src, dst, E, OUT_CH);
    bias_act_kernel<<<bNO, T, 0, stream>>>(out, bo, N, OUT_CH, 0);
}